// MemoryGraph_13898514170674
// MI455X (gfx1250) — compile-verified
//
#include <hip/hip_runtime.h>
#include <hip/hip_bf16.h>
#include <math.h>

// Problem constants (from reference)
constexpr int BATCH  = 8;
constexpr int NC     = 32;
constexpr int N      = 128;
constexpr int DN     = 64;
constexpr int ALPHA  = 4;
constexpr int HS     = 128;
constexpr int HM     = 128;
constexpr int HMOD   = 64;
constexpr int MOD_IN = 2 * DN + 5;      // 133
constexpr int MOD_OUT= N * N + N;       // 16512

// Padded LDS row strides (bank-conflict avoidance: 64 banks, pad by 4 floats)
constexpr int LDN  = N  + 4;   // 132 for 128-wide matrices
constexpr int LDD  = DN + 4;   // 68  for 64-wide matrices

// LDS layout (floats)
constexpr int OFF_W    = 0;                      // 128 x 132 = 16896
constexpr int OFF_H    = OFF_W    + N * LDN;     // 128 x 68  = 8704
constexpr int OFF_BUFA = OFF_H    + N * LDD;     // msg -> msg_new (128 x 68)
constexpr int OFF_BUFB = OFF_BUFA + N * LDD;     // received       (128 x 68)
constexpr int OFF_HID2 = OFF_BUFB + N * LDD;     // hidden -> mh   (128 x 132)
constexpr int OFF_HNEW = OFF_HID2 + N * LDN;     // h_new          (128 x 68)
constexpr int OFF_RED  = OFF_HNEW + N * LDD;     // 256 reduction slots
constexpr int OFF_MODI = OFF_RED  + 256;         // 144 (mod_input, padded)
constexpr int OFF_HIDV = OFF_MODI + 144;         // 64
constexpr int OFF_CELL = OFF_HIDV + 64;          // 64
constexpr int SM_TOT   = OFF_CELL + 64;          // ~69,200 floats (~270 KB)

typedef float v2f __attribute__((ext_vector_type(2)));
typedef float v8f __attribute__((ext_vector_type(8)));

// One 16x16 f32 WMMA tile, K-loop in steps of 4 (V_WMMA_F32_16X16X4_F32).
// A: row-major, element (m,k) at A[m*lda + k]
// B: element (k,n) at B[k*sK + n*sN]  (covers LDS k-major and global weight n-major)
__device__ __forceinline__ v8f wmma_tile(const float* A, int lda,
                                         const float* B, int sK, int sN,
                                         int K, v8f acc)
{
    const int lane = threadIdx.x & 31;
    const int rc   = lane & 15;                 // A row / B col within tile
    const int koff = (lane < 16) ? 0 : 2;       // K split across lane halves
    for (int k = 0; k < K; k += 4) {
        v2f a, b;
        a.x = A[rc * lda + k + koff];
        a.y = A[rc * lda + k + koff + 1];
        b.x = B[(k + koff)     * sK + rc * sN];
        b.y = B[(k + koff + 1) * sK + rc * sN];
        acc = __builtin_amdgcn_wmma_f32_16x16x4_f32(false, a, false, b,
                                                    (short)0, acc, false, false);
    }
    return acc;
}

__global__ void __launch_bounds__(256)
memgraph_kernel(const float* __restrict__ x,
                const float* __restrict__ h,
                const float* __restrict__ msg,
                const float* __restrict__ W,
                const float* __restrict__ decay_logit,
                const float* __restrict__ readout_drift,
                const float* __restrict__ s_mem_live,
                const float* __restrict__ s_mem_ema,
                const float* __restrict__ neuron_id,
                const float* __restrict__ state_w1,
                const float* __restrict__ state_b1,
                const float* __restrict__ state_w2,
                const float* __restrict__ state_b2,
                const float* __restrict__ msg_w1,
                const float* __restrict__ msg_b1,
                const float* __restrict__ msg_w2,
                const float* __restrict__ msg_b2,
                const float* __restrict__ inject_w,
                const float* __restrict__ inject_b,
                const float* __restrict__ mod_w1,
                const float* __restrict__ mod_b1,
                const float* __restrict__ mod_w2,
                const float* __restrict__ mod_b2,
                float* __restrict__ outR,
                float* __restrict__ outH,
                float* __restrict__ outM,
                float* __restrict__ outW,
                float* __restrict__ outD)
{
    __shared__ float smem[SM_TOT];
    float* sW    = smem + OFF_W;
    float* sH    = smem + OFF_H;
    float* sBufA = smem + OFF_BUFA;   // msg, later msg_new
    float* sBufB = smem + OFF_BUFB;   // received
    float* sHid2 = smem + OFF_HID2;   // hidden, later mh
    float* sHnew = smem + OFF_HNEW;
    float* sRed  = smem + OFF_RED;
    float* sModI = smem + OFF_MODI;
    float* sHidV = smem + OFF_HIDV;
    float* sCell = smem + OFF_CELL;

    const int tid  = threadIdx.x;
    const int wid  = tid >> 5;
    const int lane = tid & 31;
    const int rc   = lane & 15;
    const int pair = blockIdx.x;          // b*NC + c
    const int b    = pair >> 5;
    const int c    = pair & 31;

    // ================= Stage 0: stage W/msg/h/cell into LDS, |W| partial sums
    {
        const float4* Wg = (const float4*)(W + (size_t)pair * (N * N));
        float wabs = 0.f;
        #pragma unroll
        for (int i = 0; i < 16; ++i) {
            int idx  = tid + i * 256;         // float4 index over 128x128
            int row  = idx >> 5;              // 32 float4 per row
            int col4 = idx & 31;
            float4 v = Wg[idx];
            *(float4*)(sW + row * LDN + col4 * 4) = v;
            wabs += fabsf(v.x) + fabsf(v.y) + fabsf(v.z) + fabsf(v.w);
        }
        sRed[tid] = wabs;
        const float4* Mg = (const float4*)(msg + (size_t)pair * (N * DN));
        const float4* Hg = (const float4*)(h   + (size_t)pair * (N * DN));
        #pragma unroll
        for (int i = 0; i < 8; ++i) {
            int idx  = tid + i * 256;         // float4 index over 128x64
            int row  = idx >> 4;              // 16 float4 per row
            int col4 = idx & 15;
            *(float4*)(sBufA + row * LDD + col4 * 4) = Mg[idx];
            *(float4*)(sH    + row * LDD + col4 * 4) = Hg[idx];
        }
        if (tid < DN) sCell[tid] = x[(size_t)b * (NC * DN) + c * DN + tid];
    }
    __syncthreads();

    // ================= Stage 1: received = W @ msg   (128x64), 32 tiles
    for (int tt = 0; tt < 4; ++tt) {
        int t  = wid * 4 + tt;
        int m0 = (t >> 2) * 16, n0 = (t & 3) * 16;
        v8f acc = {};
        acc = wmma_tile(sW + m0 * LDN, LDN, sBufA + n0, LDD, 1, N, acc);
        #pragma unroll
        for (int r = 0; r < 8; ++r) {
            int m = m0 + ((lane < 16) ? r : r + 8);
            sBufB[m * LDD + n0 + rc] = acc[r];
        }
    }
    __syncthreads();

    // ================= Stage 1b: inject into rows 0..ALPHA-1
    {
        float a = inject_b[c * (ALPHA * DN) + tid];
        const float4* iw4 = (const float4*)(inject_w + ((size_t)c * (ALPHA * DN) + tid) * DN);
        #pragma unroll
        for (int i = 0; i < DN / 4; ++i) {
            float4 w = iw4[i];
            a += sCell[4 * i + 0] * w.x + sCell[4 * i + 1] * w.y
               + sCell[4 * i + 2] * w.z + sCell[4 * i + 3] * w.w;
        }
        sBufB[(tid >> 6) * LDD + (tid & 63)] += a;
    }
    __syncthreads();

    // ================= Stage 2: hidden = tanh(recv@w1aT + h@w1bT + b1), 64 tiles
    for (int tt = 0; tt < 8; ++tt) {
        int t  = wid * 8 + tt;
        int m0 = (t >> 3) * 16, n0 = (t & 7) * 16;
        v8f acc = {};
        acc = wmma_tile(sBufB + m0 * LDD, LDD, state_w1 + n0 * (2 * DN),      1, 2 * DN, DN, acc);
        acc = wmma_tile(sH    + m0 * LDD, LDD, state_w1 + n0 * (2 * DN) + DN, 1, 2 * DN, DN, acc);
        float bb = state_b1[n0 + rc];
        #pragma unroll
        for (int r = 0; r < 8; ++r) {
            int m = m0 + ((lane < 16) ? r : r + 8);
            sHid2[m * LDN + n0 + rc] = tanhf(acc[r] + bb);
        }
    }
    __syncthreads();

    // ================= Stage 3: cand = tanh(hidden@w2T + b2); h_new blend, 32 tiles
    for (int tt = 0; tt < 4; ++tt) {
        int t  = wid * 4 + tt;
        int m0 = (t >> 2) * 16, n0 = (t & 3) * 16;
        v8f acc = {};
        acc = wmma_tile(sHid2 + m0 * LDN, LDN, state_w2 + n0 * HS, 1, HS, HS, acc);
        float bb = state_b2[n0 + rc];
        #pragma unroll
        for (int r = 0; r < 8; ++r) {
            int m   = m0 + ((lane < 16) ? r : r + 8);
            float dl  = decay_logit[(size_t)pair * N + m];
            float dec = 1.f / (1.f + expf(-dl));
            float cd  = tanhf(acc[r] + bb);
            float hn  = dec * sH[m * LDD + n0 + rc] + (1.f - dec) * cd;
            sHnew[m * LDD + n0 + rc] = hn;
        }
    }
    __syncthreads();

    // ================= Stage 4: mh = tanh(h_new@msg_w1T + b1), 64 tiles
    for (int tt = 0; tt < 8; ++tt) {
        int t  = wid * 8 + tt;
        int m0 = (t >> 3) * 16, n0 = (t & 7) * 16;
        v8f acc = {};
        acc = wmma_tile(sHnew + m0 * LDD, LDD, msg_w1 + n0 * DN, 1, DN, DN, acc);
        float bb = msg_b1[n0 + rc];
        #pragma unroll
        for (int r = 0; r < 8; ++r) {
            int m = m0 + ((lane < 16) ? r : r + 8);
            sHid2[m * LDN + n0 + rc] = tanhf(acc[r] + bb);
        }
    }
    __syncthreads();

    // ================= Stage 5: msg_new = tanh(mh@msg_w2T + b2) + neuron_id, 32 tiles
    for (int tt = 0; tt < 4; ++tt) {
        int t  = wid * 4 + tt;
        int m0 = (t >> 2) * 16, n0 = (t & 3) * 16;
        v8f acc = {};
        acc = wmma_tile(sHid2 + m0 * LDN, LDN, msg_w2 + n0 * HM, 1, HM, HM, acc);
        float bb = msg_b2[n0 + rc];
        #pragma unroll
        for (int r = 0; r < 8; ++r) {
            int m   = m0 + ((lane < 16) ? r : r + 8);
            float v = tanhf(acc[r] + bb)
                    + neuron_id[((size_t)c * N + m) * DN + n0 + rc];
            sBufA[m * LDD + n0 + rc] = v;                       // msg_new
        }
    }
    __syncthreads();

    // ================= Stage 5b: coalesced b128 stores of h_new / msg_new
    {
        float4* outH4 = (float4*)(outH + (size_t)pair * (N * DN));
        float4* outM4 = (float4*)(outM + (size_t)pair * (N * DN));
        #pragma unroll
        for (int i = 0; i < 8; ++i) {
            int idx  = tid + i * 256;         // float4 index over 128x64
            int row  = idx >> 4;
            int col4 = idx & 15;
            outH4[idx] = *(const float4*)(sHnew + row * LDD + col4 * 4);
            outM4[idx] = *(const float4*)(sBufA + row * LDD + col4 * 4);
        }
    }

    // ================= Stage 6: stats, readout, mod_input
    // |W| mean (tree over 256 partials from stage 0)
    for (int s = 128; s > 0; s >>= 1) {
        if (tid < s) sRed[tid] += sRed[tid + s];
        __syncthreads();
    }
    float wsum = sRed[0];
    __syncthreads();
    // decay_logit mean
    sRed[tid] = (tid < N) ? decay_logit[(size_t)pair * N + tid] : 0.f;
    __syncthreads();
    for (int s = 128; s > 0; s >>= 1) {
        if (tid < s) sRed[tid] += sRed[tid + s];
        __syncthreads();
    }
    // h_mean / msg_mean over rows
    if (tid < DN) {
        float s = 0.f;
        for (int n = 0; n < N; ++n) s += sHnew[n * LDD + tid];
        sModI[tid] = s * (1.f / 128.f);
    } else if (tid < 2 * DN) {
        int d = tid - DN;
        float s = 0.f;
        for (int n = 0; n < N; ++n) s += sBufA[n * LDD + d];
        sModI[DN + d] = s * (1.f / 128.f);
    }
    if (tid == 0) {
        sModI[128] = wsum * (1.f / 16384.f);
        sModI[129] = sRed[0] * (1.f / 128.f);
        sModI[130] = readout_drift[pair];
        sModI[131] = s_mem_live[b];
        sModI[132] = s_mem_ema[b];
    }
    // readout = 0.5 * sum(msg_new rows 4..7)
    if (tid < DN) {
        float r = sBufA[4 * LDD + tid] + sBufA[5 * LDD + tid]
                + sBufA[6 * LDD + tid] + sBufA[7 * LDD + tid];
        outR[(size_t)b * (NC * DN) + c * DN + tid] = r * 0.5f;
    }
    __syncthreads();

    // ================= Stage 7: hid = tanh(mod_input @ mod_w1[c] + mod_b1[c])
    if (tid < HMOD) {
        float a = mod_b1[c * HMOD + tid];
        const float* w1p = mod_w1 + (size_t)c * MOD_IN * HMOD + tid;
        for (int i = 0; i < MOD_IN; ++i) a += sModI[i] * w1p[(size_t)i * HMOD];
        sHidV[tid] = tanhf(a);
    }
    __syncthreads();

    // ================= Stage 8: outm = hid @ mod_w2[c] + mod_b2[c]; fuse W/decay update
    // Dominant memory stream (mod_w2, ~135 MB total): b128 coalesced reads.
    {
        const float4* w2p4 = (const float4*)(mod_w2 + (size_t)c * HMOD * MOD_OUT);
        const float4* b2p4 = (const float4*)(mod_b2 + (size_t)c * MOD_OUT);
        float4* outW4 = (float4*)(outW + (size_t)pair * (N * N));
        constexpr int MO4 = MOD_OUT / 4;       // 4128 float4 per k-row
        for (int o4 = tid; o4 < MO4; o4 += 256) {
            float4 a = b2p4[o4];
            #pragma unroll 4
            for (int k = 0; k < HMOD; ++k) {
                float  hk = sHidV[k];
                float4 w  = w2p4[(size_t)k * MO4 + o4];
                a.x += hk * w.x; a.y += hk * w.y;
                a.z += hk * w.z; a.w += hk * w.w;
            }
            int o = o4 * 4;
            if (o < N * N) {
                int nr  = o >> 7;             // row of W (4 cols never straddle a row)
                int ncb = o & 127;
                const float* wrow = sW + nr * LDN;
                float4 wv;
                wv.x = (ncb + 0 == nr) ? 0.f : (wrow[ncb + 0] + a.x);
                wv.y = (ncb + 1 == nr) ? 0.f : (wrow[ncb + 1] + a.y);
                wv.z = (ncb + 2 == nr) ? 0.f : (wrow[ncb + 2] + a.z);
                wv.w = (ncb + 3 == nr) ? 0.f : (wrow[ncb + 3] + a.w);
                outW4[o4] = wv;
            } else {
                int j = o - N * N;
                const float* dlp = decay_logit + (size_t)pair * N + j;
                float4 dv;
                dv.x = dlp[0] + a.x; dv.y = dlp[1] + a.y;
                dv.z = dlp[2] + a.z; dv.w = dlp[3] + a.w;
                *(float4*)(outD + (size_t)pair * N + j) = dv;
            }
        }
    }
}

extern "C" void kernel_launch(void* const* d_in, const int* in_sizes, int n_in,
                              void* d_out, int out_size, void* d_ws, size_t ws_size,
                              hipStream_t stream) {
    (void)in_sizes; (void)n_in; (void)out_size; (void)d_ws; (void)ws_size;
    const float* x           = (const float*)d_in[0];
    const float* h           = (const float*)d_in[1];
    const float* msg         = (const float*)d_in[2];
    const float* W           = (const float*)d_in[3];
    const float* decay_logit = (const float*)d_in[4];
    const float* rdrift      = (const float*)d_in[5];
    const float* s_live      = (const float*)d_in[6];
    const float* s_ema       = (const float*)d_in[7];
    const float* neuron_id   = (const float*)d_in[8];
    const float* state_w1    = (const float*)d_in[9];
    const float* state_b1    = (const float*)d_in[10];
    const float* state_w2    = (const float*)d_in[11];
    const float* state_b2    = (const float*)d_in[12];
    const float* msg_w1      = (const float*)d_in[13];
    const float* msg_b1      = (const float*)d_in[14];
    const float* msg_w2      = (const float*)d_in[15];
    const float* msg_b2      = (const float*)d_in[16];
    const float* inject_w    = (const float*)d_in[17];
    const float* inject_b    = (const float*)d_in[18];
    const float* mod_w1      = (const float*)d_in[19];
    const float* mod_b1      = (const float*)d_in[20];
    const float* mod_w2      = (const float*)d_in[21];
    const float* mod_b2      = (const float*)d_in[22];

    float* out  = (float*)d_out;
    float* outR = out;                                   // 16384
    float* outH = out + 16384;                           // 2097152
    float* outM = out + 16384 + 2097152;                 // 2097152
    float* outW = out + 16384 + 2 * 2097152;             // 4194304
    float* outD = outW + 4194304;                        // 32768

    memgraph_kernel<<<dim3(BATCH * NC), dim3(256), 0, stream>>>(
        x, h, msg, W, decay_logit, rdrift, s_live, s_ema, neuron_id,
        state_w1, state_b1, state_w2, state_b2,
        msg_w1, msg_b1, msg_w2, msg_b2,
        inject_w, inject_b, mod_w1, mod_b1, mod_w2, mod_b2,
        outR, outH, outM, outW, outD);
}